// GraphConvPositive_51187420234089
// MI455X (gfx1250) — compile-verified
//
#include <hip/hip_runtime.h>
#include <hip/hip_bf16.h>
#include <math.h>

// Problem constants (from reference)
#define BB    4
#define NN    4096
#define EE    32768
#define ETOT  (BB * EE)        // 131072 edges
#define INC   32               // in_channels
#define OUTC  32               // out_channels
#define EDGEC 68               // 2*(IN_C + DIMP)
#define HIDN  8
#define GCOLS 256              // OUTC * HIDN

typedef __attribute__((ext_vector_type(2))) float v2f;
typedef __attribute__((ext_vector_type(8))) float v8f;

// Workspace layout (in floats)
#define WS_XF   0                       // B*N*32   = 524288
#define WS_G    524288                  // N*256    = 1048576
#define WS_CB   1572864                 // N*32     = 131072
#define WS_S    1703936                 // B*N*32   = 524288
#define WS_CNT  2228224                 // B*N      = 16384
#define WS_ZERO_LEN (524288 + 16384)    // s + cnt contiguous = 540672

// ---------------------------------------------------------------- zero s/cnt
__global__ void k_zero(float* __restrict__ p, int n) {
    int i = blockIdx.x * blockDim.x + threadIdx.x;
    if (i < n) p[i] = 0.0f;
}

// ------------------------------------------------- transpose (B,C,N)->(node,c)
__global__ void k_build_xf(const float* __restrict__ a,
                           const float* __restrict__ u,
                           float* __restrict__ xf) {
    int idx = blockIdx.x * blockDim.x + threadIdx.x;       // B*N*32 threads
    if (idx >= BB * NN * INC) return;
    int i    = idx & 31;
    int node = idx >> 5;
    int b    = node >> 12;            // node / N
    int n    = node & (NN - 1);       // node % N
    float v = (i < 16) ? a[(b * 16 + i) * NN + n]
                       : u[(b * 16 + (i - 16)) * NN + n];
    xf[idx] = v;
}

// ------------------------------------------- Cb[node,o] = sum_i x[i]*b2[i*32+o]
__global__ void k_cb(const float* __restrict__ xf,
                     const float* __restrict__ b2,
                     float* __restrict__ Cb) {
    int idx = blockIdx.x * blockDim.x + threadIdx.x;       // N*32 threads
    if (idx >= NN * OUTC) return;
    int o    = idx & 31;
    int node = idx >> 5;
    float acc = 0.0f;
    #pragma unroll
    for (int i = 0; i < INC; ++i)
        acc += xf[node * INC + i] * b2[i * OUTC + o];
    Cb[idx] = acc;
}

// --------------------------------------------------------------- G GEMM (WMMA)
// G (4096 x 256) = X (4096 x 32) @ W2view (32 x 256), fp32 WMMA 16x16x4.
// W2view[i][col] == w2_flat[i*256 + col]  (col = o*8+k), i.e. w2 reinterpreted.
__global__ void k_gemm_G(const float* __restrict__ xf,
                         const float* __restrict__ w2,
                         float* __restrict__ G) {
    int wid  = threadIdx.x >> 5;
    int tile = blockIdx.x * (blockDim.x >> 5) + wid;       // 4096 tiles
    if (tile >= (NN / 16) * (GCOLS / 16)) return;          // wave-uniform guard
    int lane = threadIdx.x & 31;
    int half = lane >> 4;
    int l    = lane & 15;
    int mt   = tile >> 4;                                  // 256 row tiles
    int nt   = tile & 15;                                  // 16 col tiles
    int row  = mt * 16 + l;
    int col  = nt * 16 + l;

    v8f acc = {};
    #pragma unroll
    for (int k0 = 0; k0 < INC; k0 += 4) {
        int ka = k0 + 2 * half;                            // ISA A/B lane layout
        v2f av, bv;
        av.x = xf[row * INC + ka];
        av.y = xf[row * INC + ka + 1];
        bv.x = w2[ka * GCOLS + col];
        bv.y = w2[(ka + 1) * GCOLS + col];
        acc = __builtin_amdgcn_wmma_f32_16x16x4_f32(
                  false, av, false, bv, (short)0, acc, false, false);
    }
    #pragma unroll
    for (int r = 0; r < 8; ++r) {                          // C layout: M=r / 8+r
        int m = mt * 16 + r + 8 * half;
        G[m * GCOLS + nt * 16 + l] = acc[r];
    }
}

// ------------------------------------------------------------- per-edge kernel
// One lane per edge: 68->8 MLP + exact GELU, then msg = Cb + h . G[src],
// scatter-add into s[dst], count into cnt[dst].
__global__ void k_edge(const float* __restrict__ xf,
                       const float* __restrict__ gridf,
                       const long long* __restrict__ ei,
                       const float* __restrict__ w1,
                       const float* __restrict__ b1,
                       const float* __restrict__ G,
                       const float* __restrict__ Cb,
                       float* __restrict__ s,
                       float* __restrict__ cnt) {
    __shared__ float sw1[HIDN * EDGEC];
    __shared__ float sb1[HIDN];
    for (int i = threadIdx.x; i < HIDN * EDGEC; i += blockDim.x) sw1[i] = w1[i];
    if (threadIdx.x < HIDN) sb1[threadIdx.x] = b1[threadIdx.x];
    __syncthreads();

    int e = blockIdx.x * blockDim.x + threadIdx.x;
    if (e >= ETOT) return;
    int base = e & (EE - 1);                 // faithful tile: 4 identical copies
    int src  = (int)ei[base];                // values in [0, N)
    int dst  = (int)ei[EE + base];

    const float* __restrict__ g  = G  + src * GCOLS;
    const float* __restrict__ cb = Cb + src * OUTC;
    __builtin_prefetch(g, 0, 0);             // global_prefetch_b8
    __builtin_prefetch(g + 64, 0, 0);

    float h[HIDN];
    #pragma unroll
    for (int k = 0; k < HIDN; ++k) h[k] = sb1[k];

    const float* xs = xf + src * INC;
    #pragma unroll
    for (int c = 0; c < INC; ++c) {
        float t = xs[c];
        #pragma unroll
        for (int k = 0; k < HIDN; ++k) h[k] += t * sw1[k * EDGEC + c];
    }
    #pragma unroll
    for (int d = 0; d < 2; ++d) {
        float t = gridf[src * 2 + d];
        #pragma unroll
        for (int k = 0; k < HIDN; ++k) h[k] += t * sw1[k * EDGEC + INC + d];
    }
    const float* xd = xf + dst * INC;
    #pragma unroll
    for (int c = 0; c < INC; ++c) {
        float t = xd[c];
        #pragma unroll
        for (int k = 0; k < HIDN; ++k) h[k] += t * sw1[k * EDGEC + 34 + c];
    }
    #pragma unroll
    for (int d = 0; d < 2; ++d) {
        float t = gridf[dst * 2 + d];
        #pragma unroll
        for (int k = 0; k < HIDN; ++k) h[k] += t * sw1[k * EDGEC + 34 + INC + d];
    }
    // exact GELU: 0.5*x*(1+erf(x/sqrt(2)))
    #pragma unroll
    for (int k = 0; k < HIDN; ++k)
        h[k] = 0.5f * h[k] * (1.0f + erff(h[k] * 0.70710678118654752f));

    #pragma unroll
    for (int o = 0; o < OUTC; ++o) {
        float m = cb[o];
        #pragma unroll
        for (int k = 0; k < HIDN; ++k) m += h[k] * g[o * HIDN + k];
        atomicAdd(&s[dst * OUTC + o], m);
    }
    atomicAdd(&cnt[dst], 1.0f);
}

// -------------------------------------------- output: agg/mean + x@root (WMMA)
__global__ void k_out(const float* __restrict__ xf,
                      const float* __restrict__ rootw,
                      const float* __restrict__ s,
                      const float* __restrict__ cnt,
                      float* __restrict__ out) {
    int wid  = threadIdx.x >> 5;
    int tile = blockIdx.x * (blockDim.x >> 5) + wid;       // 2048 tiles
    if (tile >= (BB * NN / 16) * (OUTC / 16)) return;
    int lane = threadIdx.x & 31;
    int half = lane >> 4;
    int l    = lane & 15;
    int mt   = tile >> 1;
    int nt   = tile & 1;
    int row  = mt * 16 + l;
    int col  = nt * 16 + l;

    v8f acc = {};
    #pragma unroll
    for (int k0 = 0; k0 < INC; k0 += 4) {
        int ka = k0 + 2 * half;
        v2f av, bv;
        av.x = xf[row * INC + ka];
        av.y = xf[row * INC + ka + 1];
        bv.x = rootw[ka * OUTC + col];
        bv.y = rootw[(ka + 1) * OUTC + col];
        acc = __builtin_amdgcn_wmma_f32_16x16x4_f32(
                  false, av, false, bv, (short)0, acc, false, false);
    }
    #pragma unroll
    for (int r = 0; r < 8; ++r) {
        int node = mt * 16 + r + 8 * half;
        int o    = nt * 16 + l;
        float c  = cnt[node];
        float agg = s[node * OUTC + o] / fmaxf(c, 1.0f);
        int b = node >> 12;
        int n = node & (NN - 1);
        out[(b * OUTC + o) * NN + n] = acc[r] + agg;       // (B, OUT_C, N)
    }
}

// ----------------------------------------------------------------------------
extern "C" void kernel_launch(void* const* d_in, const int* in_sizes, int n_in,
                              void* d_out, int out_size, void* d_ws, size_t ws_size,
                              hipStream_t stream) {
    const float*     a     = (const float*)d_in[0];
    const float*     u     = (const float*)d_in[1];
    const float*     gridf = (const float*)d_in[2];
    const long long* ei    = (const long long*)d_in[3];   // int64 per reference
    const float*     w1    = (const float*)d_in[4];
    const float*     b1    = (const float*)d_in[5];
    const float*     w2    = (const float*)d_in[6];
    const float*     b2    = (const float*)d_in[7];
    const float*     root  = (const float*)d_in[8];
    float* out = (float*)d_out;

    float* ws  = (float*)d_ws;
    float* xf  = ws + WS_XF;
    float* G   = ws + WS_G;
    float* Cb  = ws + WS_CB;
    float* s   = ws + WS_S;
    float* cnt = ws + WS_CNT;

    // 1) zero scatter buffers (s + cnt are contiguous)
    k_zero<<<(WS_ZERO_LEN + 255) / 256, 256, 0, stream>>>(s, WS_ZERO_LEN);
    // 2) node features (B*N, 32)
    k_build_xf<<<(BB * NN * INC + 255) / 256, 256, 0, stream>>>(a, u, xf);
    // 3) per-node factored second layer: G = X @ W2view  (fp32 WMMA)
    k_gemm_G<<<512, 256, 0, stream>>>(xf, w2, G);
    // 4) bias contraction Cb = X @ b2view
    k_cb<<<(NN * OUTC + 255) / 256, 256, 0, stream>>>(xf, b2, Cb);
    // 5) edges: MLP + GELU + h.G contraction + atomic scatter
    k_edge<<<ETOT / 256, 256, 0, stream>>>(xf, gridf, ei, w1, b1, G, Cb, s, cnt);
    // 6) mean + root GEMM (fp32 WMMA), write (B, OUT_C, N)
    k_out<<<256, 256, 0, stream>>>(xf, root, s, cnt, out);
}